// ReactionTerm_88390426951972
// MI455X (gfx1250) — compile-verified
//
#include <hip/hip_runtime.h>
#include <stdint.h>

// Problem constants (match the reference).
#define BATCH 1024
#define NSPEC 2048
#define R1CNT 20000
#define R2CNT 40000
#define GB    2        // batch rows per block (amortizes shared index reads)
#define TPB   512      // 16 wave32s per block; 512 blocks total

typedef float v4f __attribute__((ext_vector_type(4)));
typedef int   v4i __attribute__((ext_vector_type(4)));

// --- gfx1250 async global->LDS copy (ASYNCcnt-tracked, bypasses VGPR data path) ---
__device__ __forceinline__ void async_ld_b128(const void* gptr, const void* lptr) {
    // Low 32 bits of a flat LDS pointer are the wave-relative LDS byte address
    // (CDNA5 aperture rule).
    unsigned lds_addr = (unsigned)(uint64_t)(uintptr_t)lptr;
    uint64_t  gaddr   = (uint64_t)(uintptr_t)gptr;
    asm volatile("global_load_async_to_lds_b128 %0, %1, off"
                 :: "v"(lds_addr), "v"(gaddr) : "memory");
}

__device__ __forceinline__ void wait_async0() {
#if defined(__has_builtin) && __has_builtin(__builtin_amdgcn_s_wait_asynccnt)
    __builtin_amdgcn_s_wait_asynccnt(0);
#else
    asm volatile("s_wait_asynccnt 0" ::: "memory");
#endif
}

__global__ __launch_bounds__(TPB)
void ReactionTerm_88390426951972_kernel(const float* __restrict__ y_in,
                                        const float* __restrict__ rates1,
                                        const float* __restrict__ rates2,
                                        const int*   __restrict__ i1r,
                                        const int*   __restrict__ i1p,
                                        const int*   __restrict__ i2r,   // [R2][2]
                                        const int*   __restrict__ i2p,
                                        float*       __restrict__ y_out) {
    __shared__ float sy[GB][NSPEC];    // species abundances (GB rows, 16 KB)
    __shared__ float sacc[GB][NSPEC];  // scatter accumulators       (16 KB)

    const int tid = threadIdx.x;
    const int b0  = blockIdx.x * GB;

    // ---- Async fill of the GB species rows into LDS: GB*512 chunks of 16B ----
    for (int c = tid; c < GB * (NSPEC / 4); c += TPB) {
        const int g = c >> 9;             // NSPEC/4 == 512 chunks per row
        const int k = (c & 511) * 4;
        async_ld_b128(&y_in[(size_t)(b0 + g) * NSPEC + k], &sy[g][k]);
    }
    // Zero the accumulators while the async copies are in flight.
    for (int c = tid; c < GB * (NSPEC / 4); c += TPB) {
        const int g = c >> 9;
        const int k = (c & 511) * 4;
        *(v4f*)&sacc[g][k] = (v4f){0.f, 0.f, 0.f, 0.f};
    }
    wait_async0();
    __syncthreads();

    // ---- First-order reactions: 4 reactions per thread-iteration ----
    // Indices: regular (cached, shared across all 512 blocks -> L2 resident).
    // Rates: non-temporal (240 MB read-once stream; keep it out of L2).
    const int nv1 = R1CNT / 4;
    for (int v = tid; v < nv1; v += TPB) {
        const v4i ir = *((const v4i*)i1r + v);
        const v4i ip = *((const v4i*)i1p + v);
#pragma unroll
        for (int g = 0; g < GB; ++g) {
            const v4f rt = __builtin_nontemporal_load(
                (const v4f*)(rates1 + (size_t)(b0 + g) * R1CNT) + v);
            atomicAdd(&sacc[g][ip.x], sy[g][ir.x] * rt.x);   // ds_add_f32
            atomicAdd(&sacc[g][ip.y], sy[g][ir.y] * rt.y);
            atomicAdd(&sacc[g][ip.z], sy[g][ir.z] * rt.z);
            atomicAdd(&sacc[g][ip.w], sy[g][ir.w] * rt.w);
        }
    }

    // ---- Second-order reactions: 4 reactions (8 reactant indices) per iteration ----
    const int nv2 = R2CNT / 4;
    for (int v = tid; v < nv2; v += TPB) {
        const v4i a  = *((const v4i*)i2r + 2 * v);       // (r0a,r0b,r1a,r1b)
        const v4i bq = *((const v4i*)i2r + 2 * v + 1);   // (r2a,r2b,r3a,r3b)
        const v4i ip = *((const v4i*)i2p + v);
#pragma unroll
        for (int g = 0; g < GB; ++g) {
            const v4f rt = __builtin_nontemporal_load(
                (const v4f*)(rates2 + (size_t)(b0 + g) * R2CNT) + v);
            atomicAdd(&sacc[g][ip.x], sy[g][a.x]  * sy[g][a.y]  * rt.x);
            atomicAdd(&sacc[g][ip.y], sy[g][a.z]  * sy[g][a.w]  * rt.y);
            atomicAdd(&sacc[g][ip.z], sy[g][bq.x] * sy[g][bq.y] * rt.z);
            atomicAdd(&sacc[g][ip.w], sy[g][bq.z] * sy[g][bq.w] * rt.w);
        }
    }
    __syncthreads();

    // ---- Coalesced non-temporal b128 writeback of the result rows ----
    for (int c = tid; c < GB * (NSPEC / 4); c += TPB) {
        const int g = c >> 9;
        const int k = (c & 511) * 4;
        const v4f r = *(const v4f*)&sacc[g][k];
        __builtin_nontemporal_store(r, (v4f*)&y_out[(size_t)(b0 + g) * NSPEC + k]);
    }
}

extern "C" void kernel_launch(void* const* d_in, const int* in_sizes, int n_in,
                              void* d_out, int out_size, void* d_ws, size_t ws_size,
                              hipStream_t stream) {
    (void)in_sizes; (void)n_in; (void)d_ws; (void)ws_size; (void)out_size;
    const float* y_in   = (const float*)d_in[0];
    const float* rates1 = (const float*)d_in[1];
    const float* rates2 = (const float*)d_in[2];
    const int*   i1r    = (const int*)d_in[3];
    const int*   i1p    = (const int*)d_in[4];
    const int*   i2r    = (const int*)d_in[5];
    const int*   i2p    = (const int*)d_in[6];
    float*       y_out  = (float*)d_out;

    dim3 grid(BATCH / GB);   // 512 blocks
    dim3 block(TPB);         // 512 threads = 16 wave32s
    ReactionTerm_88390426951972_kernel<<<grid, block, 0, stream>>>(
        y_in, rates1, rates2, i1r, i1p, i2r, i2p, y_out);
}